// GINRegression_12781822673318
// MI455X (gfx1250) — compile-verified
//
#include <hip/hip_runtime.h>

#define N_NODES    100000
#define N_EDGES    1600000
#define HIDDEN     64
#define NUM_GRAPHS 256

typedef __attribute__((ext_vector_type(2))) float v2f;
typedef __attribute__((ext_vector_type(8))) float v8f;
typedef __attribute__((ext_vector_type(4))) int   v4i;

#if defined(__HIP_DEVICE_COMPILE__) && __has_builtin(__builtin_amdgcn_global_load_async_to_lds_b128)
#define HAVE_ASYNC_LDS 1
typedef __attribute__((address_space(1))) v4i* gptr_v4i;   // global
typedef __attribute__((address_space(3))) v4i* lptr_v4i;   // LDS
#else
#define HAVE_ASYNC_LDS 0
#endif

__device__ __forceinline__ void wait_asynccnt0() {
#if defined(__HIP_DEVICE_COMPILE__)
#if __has_builtin(__builtin_amdgcn_s_wait_asynccnt)
    __builtin_amdgcn_s_wait_asynccnt(0);
#else
    asm volatile("s_wait_asynccnt 0" ::: "memory");
#endif
#endif
}

// ---------------------------------------------------------------- zero fill
__global__ void zero_f32(float* __restrict__ p, int n) {
    int i = blockIdx.x * blockDim.x + threadIdx.x;
    if (i < n) p[i] = 0.0f;
}

// ------------------------------------------------- edge gather + scatter-add
// 64 threads per edge: lane f handles feature f. Gather x[src] and atomically
// accumulate into agg[dst]. Both arrays are L2-resident (25.6 MB each).
__global__ void edge_scatter(const float* __restrict__ h,
                             const long long* __restrict__ src,
                             const long long* __restrict__ dst,
                             float* __restrict__ agg) {
    long long gid = (long long)blockIdx.x * blockDim.x + threadIdx.x;
    if (gid >= (long long)N_EDGES * HIDDEN) return;
    int e = (int)(gid >> 6);
    int f = (int)(gid & 63);
    int s = (int)src[e];
    int d = (int)dst[e];
    atomicAdd(&agg[(long long)d * HIDDEN + f], h[(long long)s * HIDDEN + f]);
}

// ------------------------------------------------------- fused GIN MLP layer
// hout = act( relu( (agg + xin) @ W1 + b1 ) @ W2 + b2 )
// 128 threads = 4 waves; each wave owns a 16-row tile; block covers 64 rows.
// Uses V_WMMA_F32_16X16X4_F32 (fp32 fidelity; problem is memory-bound anyway).
// Weights are staged via async global->LDS copies (ASYNCcnt) when available.
__global__ __launch_bounds__(128)
void gin_mlp(const float* __restrict__ agg,
             const float* __restrict__ xin,
             const float* __restrict__ W1, const float* __restrict__ b1,
             const float* __restrict__ W2, const float* __restrict__ b2,
             float* __restrict__ hout, int relu_out) {
    __shared__ __align__(16) float h0s[64 * 64];   // 16 KB: (agg+x) tile, reused for h1
    __shared__ __align__(16) float w1s[64 * 64];   // 16 KB
    __shared__ __align__(16) float w2s[64 * 64];   // 16 KB
    __shared__ float b1s[64];
    __shared__ float b2s[64];

    const int tid  = threadIdx.x;
    const int row0 = blockIdx.x * 64;

#if HAVE_ASYNC_LDS
    // Async DMA of the two 16KB weight tiles straight into LDS (no VGPR round
    // trip); overlaps with the VALU-carrying h0 staging below.
    for (int i = tid; i < 1024; i += 128) {
        __builtin_amdgcn_global_load_async_to_lds_b128(
            (gptr_v4i)((const float4*)W1 + i),
            (lptr_v4i)((float4*)w1s + i), 0, 0);
        __builtin_amdgcn_global_load_async_to_lds_b128(
            (gptr_v4i)((const float4*)W2 + i),
            (lptr_v4i)((float4*)w2s + i), 0, 0);
    }
#endif

    // Cooperative staging of h0 = agg + x (float4 vectorized, coalesced)
    for (int i = tid; i < 1024; i += 128) {
        int r = i >> 4, c4 = i & 15;
        int row = row0 + r;
        int rc  = row < N_NODES ? row : (N_NODES - 1);   // clamp keeps EXEC full
        float4 a = ((const float4*)(agg + (size_t)rc * 64))[c4];
        float4 x = ((const float4*)(xin + (size_t)rc * 64))[c4];
        float4 hh; hh.x = a.x + x.x; hh.y = a.y + x.y; hh.z = a.z + x.z; hh.w = a.w + x.w;
        ((float4*)h0s)[i] = hh;
    }
#if !HAVE_ASYNC_LDS
    for (int i = tid; i < 1024; i += 128) {
        ((float4*)w1s)[i] = ((const float4*)W1)[i];
        ((float4*)w2s)[i] = ((const float4*)W2)[i];
    }
#endif
    if (tid < 64) { b1s[tid] = b1[tid]; b2s[tid] = b2[tid]; }
#if HAVE_ASYNC_LDS
    wait_asynccnt0();
#endif
    __syncthreads();

    const int wave = tid >> 5;        // 0..3
    const int lane = tid & 31;
    const int m    = lane & 15;       // A: row-in-tile; B/C: column
    const int hi   = lane >> 4;       // half-wave select
    const int koff = hi * 2;          // fp32 A layout: lanes 16-31 hold K+2,K+3
    const int rloc = wave * 16 + m;   // wave-private LDS row for A fragments

    v8f acc[4];

    // ---- stage 1: h1 = relu(h0 @ W1 + b1); bias pre-splat into C (depends on N only)
    for (int t = 0; t < 4; ++t) {
        float bv = b1s[t * 16 + m];
        for (int j = 0; j < 8; ++j) acc[t][j] = bv;
    }
    for (int k = 0; k < 16; ++k) {
        int c = k * 4 + koff;
        v2f a = *(const v2f*)&h0s[rloc * 64 + c];     // A[m][c], A[m][c+1]
        for (int t = 0; t < 4; ++t) {
            int col = t * 16 + m;
            v2f b;
            b.x = w1s[c * 64 + col];                  // B[K=c][col]
            b.y = w1s[(c + 1) * 64 + col];            // B[K=c+1][col]
            acc[t] = __builtin_amdgcn_wmma_f32_16x16x4_f32(
                false, a, false, b, (short)0, acc[t], false, false);
        }
    }
    for (int t = 0; t < 4; ++t)
        for (int j = 0; j < 8; ++j) acc[t][j] = fmaxf(acc[t][j], 0.0f);

    // ---- transpose h1 through wave-private LDS slice (D layout -> A layout)
    for (int t = 0; t < 4; ++t) {
        int col   = t * 16 + m;
        int mbase = wave * 16 + hi * 8;
        for (int j = 0; j < 8; ++j)
            h0s[(mbase + j) * 64 + col] = acc[t][j];
    }
    __syncthreads();

    // ---- stage 2: h2 = h1 @ W2 + b2
    for (int t = 0; t < 4; ++t) {
        float bv = b2s[t * 16 + m];
        for (int j = 0; j < 8; ++j) acc[t][j] = bv;
    }
    for (int k = 0; k < 16; ++k) {
        int c = k * 4 + koff;
        v2f a = *(const v2f*)&h0s[rloc * 64 + c];
        for (int t = 0; t < 4; ++t) {
            int col = t * 16 + m;
            v2f b;
            b.x = w2s[c * 64 + col];
            b.y = w2s[(c + 1) * 64 + col];
            acc[t] = __builtin_amdgcn_wmma_f32_16x16x4_f32(
                false, a, false, b, (short)0, acc[t], false, false);
        }
    }
    if (relu_out)
        for (int t = 0; t < 4; ++t)
            for (int j = 0; j < 8; ++j) acc[t][j] = fmaxf(acc[t][j], 0.0f);

    // ---- store D tiles (guarded for the ragged last block)
    for (int t = 0; t < 4; ++t) {
        int col   = t * 16 + m;
        int mbase = wave * 16 + hi * 8;
        for (int j = 0; j < 8; ++j) {
            int row = row0 + mbase + j;
            if (row < N_NODES) hout[(size_t)row * 64 + col] = acc[t][j];
        }
    }
}

// --------------------------------- last GIN layer (64->1->1) + pooled sums
__global__ void last_layer(const float* __restrict__ agg,
                           const float* __restrict__ h,
                           const float* __restrict__ W1l,
                           const float* __restrict__ b1l,
                           const float* __restrict__ W2l,
                           const float* __restrict__ b2l,
                           const long long* __restrict__ batch,
                           float* __restrict__ sums,
                           float* __restrict__ cnt) {
    int n = blockIdx.x * blockDim.x + threadIdx.x;
    if (n >= N_NODES) return;
    const float4* ap = (const float4*)(agg + (size_t)n * 64);
    const float4* hp = (const float4*)(h   + (size_t)n * 64);
    const float4* wp = (const float4*)W1l;
    float z = 0.0f;
    for (int i = 0; i < 16; ++i) {
        float4 a = ap[i], x = hp[i], w = wp[i];
        z += (a.x + x.x) * w.x + (a.y + x.y) * w.y
           + (a.z + x.z) * w.z + (a.w + x.w) * w.w;
    }
    z = fmaxf(z + b1l[0], 0.0f);
    float val = z * W2l[0] + b2l[0];
    int g = (int)batch[n];
    atomicAdd(&sums[g], val);
    atomicAdd(&cnt[g], 1.0f);
}

__global__ void finalize_pool(const float* __restrict__ sums,
                              const float* __restrict__ cnt,
                              float* __restrict__ out) {
    int g = blockIdx.x * blockDim.x + threadIdx.x;
    if (g < NUM_GRAPHS) out[g] = sums[g] / fmaxf(cnt[g], 1.0f);
}

// ---------------------------------------------------------------- launcher
extern "C" void kernel_launch(void* const* d_in, const int* in_sizes, int n_in,
                              void* d_out, int out_size, void* d_ws, size_t ws_size,
                              hipStream_t stream) {
    (void)in_sizes; (void)n_in; (void)out_size; (void)ws_size;
    const float*     x     = (const float*)d_in[0];
    const long long* ei    = (const long long*)d_in[1];
    const long long* src   = ei;                 // edge_index[0]
    const long long* dst   = ei + N_EDGES;       // edge_index[1]
    const long long* batch = (const long long*)d_in[2];
    const float* W1  = (const float*)d_in[3];
    const float* b1  = (const float*)d_in[4];
    const float* W2  = (const float*)d_in[5];
    const float* b2  = (const float*)d_in[6];
    const float* W1l = (const float*)d_in[7];
    const float* b1l = (const float*)d_in[8];
    const float* W2l = (const float*)d_in[9];
    const float* b2l = (const float*)d_in[10];
    float* out = (float*)d_out;

    float* ws = (float*)d_ws;
    const size_t NF = (size_t)N_NODES * HIDDEN;  // 6.4M floats
    float* agg  = ws;
    float* bufA = ws + NF;
    float* bufB = ws + 2 * NF;
    float* sums = ws + 3 * NF;                   // 256
    float* cnt  = sums + NUM_GRAPHS;             // 256

    const int zgrid = (int)((NF + 255) / 256);
    const int sgrid = (int)(((long long)N_EDGES * HIDDEN + 255) / 256);
    const int mgrid = (N_NODES + 63) / 64;
    const int lgrid = (N_NODES + 255) / 256;

    float* bufs[2] = { bufA, bufB };
    const float* hin = x;
    for (int layer = 0; layer < 4; ++layer) {
        zero_f32<<<zgrid, 256, 0, stream>>>(agg, (int)NF);
        edge_scatter<<<sgrid, 256, 0, stream>>>(hin, src, dst, agg);
        float* hout = bufs[layer & 1];
        gin_mlp<<<mgrid, 128, 0, stream>>>(agg, hin,
                                           W1 + (size_t)layer * HIDDEN * HIDDEN,
                                           b1 + (size_t)layer * HIDDEN,
                                           W2 + (size_t)layer * HIDDEN * HIDDEN,
                                           b2 + (size_t)layer * HIDDEN,
                                           hout, 1);
        hin = hout;
    }
    zero_f32<<<zgrid, 256, 0, stream>>>(agg, (int)NF);
    zero_f32<<<2, 256, 0, stream>>>(sums, 2 * NUM_GRAPHS);
    edge_scatter<<<sgrid, 256, 0, stream>>>(hin, src, dst, agg);
    last_layer<<<lgrid, 256, 0, stream>>>(agg, hin, W1l, b1l, W2l, b2l,
                                          batch, sums, cnt);
    finalize_pool<<<1, 256, 0, stream>>>(sums, cnt, out);
}